// WTConv2d_84963043050080
// MI455X (gfx1250) — compile-verified
//
#include <hip/hip_runtime.h>
#include <hip/hip_bf16.h>
#include <stdint.h>

// ---------------------------------------------------------------------------
// CDNA5 async global->LDS staging helpers (gfx1250 ASYNCcnt path)
// Builtin signature (probe-confirmed round 1/2): (as1 v2i*, as3 v2i*, Ii, Ii)
// ---------------------------------------------------------------------------
typedef int async_v2i __attribute__((vector_size(8)));

__device__ __forceinline__ void async_copy_b64(float* dst_lds, const float* src_glb) {
#if defined(__gfx1250__) && __has_builtin(__builtin_amdgcn_global_load_async_to_lds_b64)
    __builtin_amdgcn_global_load_async_to_lds_b64(
        (__attribute__((address_space(1))) async_v2i*)(src_glb),
        (__attribute__((address_space(3))) async_v2i*)(dst_lds),
        /*offset=*/0, /*cpol=*/0);
#else
    dst_lds[0] = src_glb[0];
    dst_lds[1] = src_glb[1];
#endif
}

__device__ __forceinline__ void wait_async_lds() {
#if defined(__gfx1250__) && __has_builtin(__builtin_amdgcn_global_load_async_to_lds_b64)
#if __has_builtin(__builtin_amdgcn_s_wait_asynccnt)
    __builtin_amdgcn_s_wait_asynccnt(0);
#else
    asm volatile("s_wait_asynccnt 0" ::: "memory");
#endif
#endif
}

// ---------------------------------------------------------------------------
// CDNA5 WMMA: V_WMMA_F32_16X16X4_F32 for the Haar butterfly
//   D(16x16) = A(16x4) x B(4x16) + C
//   A rows 0..3 = 0.5 * Haar matrix (constant), B columns = 2x2 input blocks.
// ---------------------------------------------------------------------------
typedef float wm_v2f __attribute__((ext_vector_type(2)));
typedef float wm_v8f __attribute__((ext_vector_type(8)));

#if defined(__gfx1250__) && __has_builtin(__builtin_amdgcn_wmma_f32_16x16x4_f32)
#define HAVE_WMMA_F32X4 1
#else
#define HAVE_WMMA_F32X4 0
#endif

// ---------------------------------------------------------------------------
// Kernel 1: fused haar2d + 5x5 depthwise conv (+scale) for one wavelet level.
//   cur      : (B,C,2h,2w) level input
//   wv       : (4C,25) level conv weights, wsc: (4C) level scales
//   lc       : (B,C,4,h,w) conv'd coefficients (output)
//   nextcur  : (B,C,h,w) pre-conv LL for next level (optional)
// Tile: 14x14 coefficient outputs; 18x18 coeffs (conv halo 2) from a 36x36
// input tile staged through the async global->LDS path.
// ---------------------------------------------------------------------------
#define TILE   14
#define CTILE  (TILE + 4)    // 18
#define XTILE  (2 * CTILE)   // 36

__global__ __launch_bounds__(256)
void k_haar_dwconv(const float* __restrict__ cur,
                   const float* __restrict__ wv,
                   const float* __restrict__ wsc,
                   float* __restrict__ lc,
                   float* __restrict__ nextcur,
                   int C, int hc, int wc, int tilesx)
{
    __shared__ __align__(16) float sCur[XTILE][XTILE];
    __shared__ float sHaar[4][CTILE][CTILE];
    __shared__ float sW[4][25];
    __shared__ float sS[4];

    const int b   = blockIdx.z;
    const int c   = blockIdx.y;
    const int tyb = blockIdx.x / tilesx;
    const int txb = blockIdx.x % tilesx;
    const int t   = threadIdx.x;
    const int Hc  = 2 * hc, Wc = 2 * wc;
    const int y0  = tyb * TILE, x0 = txb * TILE;     // coeff-space tile origin
    const int gr0 = 2 * y0 - 4, gc0 = 2 * x0 - 4;    // input-space tile origin

    const float* curp = cur + (size_t)(b * C + c) * Hc * Wc;

    if (t < 100) sW[t / 25][t % 25] = wv[(size_t)(c * 4 + t / 25) * 25 + (t % 25)];
    if (t < 4)   sS[t] = wsc[c * 4 + t];

    const bool interior = (gr0 >= 0) && (gc0 >= 0) &&
                          (gr0 + XTILE <= Hc) && (gc0 + XTILE <= Wc);
    if (interior) {
        // 36 rows x 18 b64 chunks = 648 async copies (8B-aligned both sides)
        for (int idx = t; idx < XTILE * (XTILE / 2); idx += 256) {
            const int row = idx / (XTILE / 2);
            const int ch  = idx % (XTILE / 2);
            async_copy_b64(&sCur[row][ch * 2],
                           curp + (size_t)(gr0 + row) * Wc + gc0 + ch * 2);
        }
        wait_async_lds();
    } else {
        for (int idx = t; idx < XTILE * XTILE; idx += 256) {
            const int row = idx / XTILE, col = idx % XTILE;
            const int gr = gr0 + row, gc = gc0 + col;
            float v = 0.f;
            if (gr >= 0 && gr < Hc && gc >= 0 && gc < Wc)
                v = curp[(size_t)gr * Wc + gc];
            sCur[row][col] = v;
        }
    }
    __syncthreads();

    // ---- Haar butterfly over 18x18 coefficient positions -------------------
#if HAVE_WMMA_F32X4
    {
        const int lane = t & 31;
        const int wvid = t >> 5;          // wave 0..7
        const int m    = lane & 15;       // M row (lanes 0-15) / same for hi half
        const bool hi  = lane >= 16;      // hi half supplies K=2,3

        // A(16x4): rows 0..3 = 0.5*H4, rows 4..15 = 0
        //   H[m][k] sign = ((k&1)&&(m&1) ? -1:1) * ((k>=2)&&(m>=2) ? -1:1)
        const int k0 = hi ? 2 : 0;
        const int k1 = k0 + 1;
        float s0 = (((k0 & 1) && (m & 1)) ? -0.5f : 0.5f) * (((k0 >= 2) && (m >= 2)) ? -1.f : 1.f);
        float s1 = (((k1 & 1) && (m & 1)) ? -0.5f : 0.5f) * (((k1 >= 2) && (m >= 2)) ? -1.f : 1.f);
        wm_v2f A;
        A.x = (m < 4) ? s0 : 0.f;
        A.y = (m < 4) ? s1 : 0.f;

        const int nPos    = CTILE * CTILE;          // 324
        const int nChunks = (nPos + 15) / 16;       // 21 (wave-uniform loop)
        for (int chunk = wvid; chunk < nChunks; chunk += 8) {
            const int p = chunk * 16 + m;           // coefficient position
            const int i = p / CTILE, j = p % CTILE;
            const int gy = y0 - 2 + i, gx = x0 - 2 + j;
            const bool valid = (p < nPos) && (gy >= 0) && (gy < hc) &&
                               (gx >= 0) && (gx < wc);
            // B(4x16): col n = position; rows k = {a,b,c,d} of the 2x2 block
            float e0 = 0.f, e1 = 0.f;
            if (valid) {
                const int rr = 2 * i + (hi ? 1 : 0);
                e0 = sCur[rr][2 * j];               // k0 row (a or c)
                e1 = sCur[rr][2 * j + 1];           // k1 row (b or d)
            }
            wm_v2f Bv; Bv.x = e0; Bv.y = e1;
            wm_v8f acc = {};
            acc = __builtin_amdgcn_wmma_f32_16x16x4_f32(
                false, A, false, Bv, (short)0, acc, false, false);
            // D rows 0..3 live in acc[0..3] of lanes 0..15
            if (!hi && p < nPos) {
                sHaar[0][i][j] = acc[0];
                sHaar[1][i][j] = acc[1];
                sHaar[2][i][j] = acc[2];
                sHaar[3][i][j] = acc[3];
            }
        }
    }
#else
    for (int idx = t; idx < CTILE * CTILE; idx += 256) {
        const int i = idx / CTILE, j = idx % CTILE;
        const int gy = y0 - 2 + i, gx = x0 - 2 + j;
        float ll = 0.f, lh = 0.f, hl = 0.f, hh = 0.f;
        if (gy >= 0 && gy < hc && gx >= 0 && gx < wc) {
            const float a  = sCur[2 * i    ][2 * j    ];
            const float bb = sCur[2 * i    ][2 * j + 1];
            const float cc = sCur[2 * i + 1][2 * j    ];
            const float dd = sCur[2 * i + 1][2 * j + 1];
            ll = (a + bb + cc + dd) * 0.5f;
            lh = (a - bb + cc - dd) * 0.5f;
            hl = (a + bb - cc - dd) * 0.5f;
            hh = (a - bb - cc + dd) * 0.5f;
        }
        sHaar[0][i][j] = ll; sHaar[1][i][j] = lh;
        sHaar[2][i][j] = hl; sHaar[3][i][j] = hh;
    }
#endif
    __syncthreads();

    if (t < TILE * TILE) {
        const int ty = t / TILE, tx = t % TILE;
        const int gy = y0 + ty, gx = x0 + tx;
        float* lcp = lc + (size_t)(b * C + c) * 4 * hc * wc;
#pragma unroll
        for (int s = 0; s < 4; ++s) {
            float acc = 0.f;
#pragma unroll
            for (int ki = 0; ki < 5; ++ki)
#pragma unroll
                for (int kj = 0; kj < 5; ++kj)
                    acc = fmaf(sHaar[s][ty + ki][tx + kj], sW[s][ki * 5 + kj], acc);
            lcp[((size_t)s * hc + gy) * wc + gx] = acc * sS[s];
        }
        if (nextcur)
            nextcur[((size_t)(b * C + c) * hc + gy) * wc + gx] = sHaar[0][ty + 2][tx + 2];
    }
}

// ---------------------------------------------------------------------------
// Kernel 2: inverse Haar for one level (streaming).
// ---------------------------------------------------------------------------
__global__ __launch_bounds__(256)
void k_ihaar(const float* __restrict__ lc,
             const float* __restrict__ llin,
             float* __restrict__ out,
             int n, int h, int w)
{
    const int i = blockIdx.x * 256 + threadIdx.x;
    if (i >= n) return;
    const int x = i % w;
    const int y = (i / w) % h;
    const size_t bc = (size_t)i / ((size_t)w * h);
    const size_t plane = (size_t)h * w;
    const float* p = lc + bc * 4 * plane + (size_t)y * w + x;

    float ll = p[0] + (llin ? llin[i] : 0.f);
    const float lh = p[plane], hl = p[2 * plane], hh = p[3 * plane];
    const float a  = (ll + lh + hl + hh) * 0.5f;
    const float b2 = (ll - lh + hl - hh) * 0.5f;
    const float c2 = (ll + lh - hl - hh) * 0.5f;
    const float d2 = (ll - lh - hl + hh) * 0.5f;

    float* o = out + bc * plane * 4 + (size_t)(2 * y) * (2 * w) + 2 * x;
    *reinterpret_cast<float2*>(o)         = make_float2(a, b2);
    *reinterpret_cast<float2*>(o + 2 * w) = make_float2(c2, d2);
}

// ---------------------------------------------------------------------------
// Kernel 3: fused base 5x5 dwconv (+scale+bias) + level-0 inverse Haar + add.
// ---------------------------------------------------------------------------
__global__ __launch_bounds__(256)
void k_final(const float* __restrict__ x,
             const float* __restrict__ bw,   // (C,25)
             const float* __restrict__ bb,   // (C)
             const float* __restrict__ bs,   // (C)
             const float* __restrict__ lc0,  // (B,C,4,112,112)
             const float* __restrict__ ll,   // (B,C,112,112)
             float* __restrict__ out,
             int C, int tilesx)
{
    const int H = 224, W = 224, hc = 112, wc = 112;
    __shared__ __align__(16) float sX[TILE + 4][TILE + 4];  // 18x18
    __shared__ float sCo[5][8][8];
    __shared__ float sW[25];
    __shared__ float sSB[2];

    const int b   = blockIdx.z;
    const int c   = blockIdx.y;
    const int tyb = blockIdx.x / tilesx;
    const int txb = blockIdx.x % tilesx;
    const int t   = threadIdx.x;
    const int y0  = tyb * TILE, x0 = txb * TILE;   // always even
    const int cy0 = y0 >> 1,   cx0 = x0 >> 1;

    const float* xp  = x   + (size_t)(b * C + c) * H * W;
    const float* lcp = lc0 + (size_t)(b * C + c) * 4 * hc * wc;
    const float* llp = ll  + (size_t)(b * C + c) * hc * wc;

    if (t < 25)       sW[t]   = bw[(size_t)c * 25 + t];
    else if (t == 25) sSB[0]  = bs[c];
    else if (t == 26) sSB[1]  = bb[c];

    // Stage 5 coefficient tiles (4 subbands + incoming LL), 8x8 each, guarded.
    for (int idx = t; idx < 5 * 64; idx += 256) {
        const int s = idx / 64, rr = (idx / 8) % 8, cc2 = idx % 8;
        const int gy = cy0 + rr, gx = cx0 + cc2;
        float v = 0.f;
        if (gy < hc && gx < wc)
            v = (s < 4) ? lcp[((size_t)s * hc + gy) * wc + gx]
                        : llp[(size_t)gy * wc + gx];
        sCo[s][rr][cc2] = v;
    }

    // Stage x tile 18x18 (conv halo 2) — async for interior tiles.
    const int gr0 = y0 - 2, gc0 = x0 - 2;
    const bool interior = gr0 >= 0 && gc0 >= 0 && gr0 + 18 <= H && gc0 + 18 <= W;
    if (interior) {
        for (int idx = t; idx < 18 * 9; idx += 256) {
            const int row = idx / 9, ch = idx % 9;
            async_copy_b64(&sX[row][ch * 2],
                           xp + (size_t)(gr0 + row) * W + gc0 + ch * 2);
        }
        wait_async_lds();
    } else {
        for (int idx = t; idx < 18 * 18; idx += 256) {
            const int row = idx / 18, col = idx % 18;
            const int gr = gr0 + row, gc = gc0 + col;
            sX[row][col] = (gr >= 0 && gr < H && gc >= 0 && gc < W)
                               ? xp[(size_t)gr * W + gc] : 0.f;
        }
    }
    __syncthreads();

    if (t < TILE * TILE) {
        const int ty = t / TILE, tx = t % TILE;
        const int y = y0 + ty, xg = x0 + tx;
        float acc = 0.f;
#pragma unroll
        for (int ki = 0; ki < 5; ++ki)
#pragma unroll
            for (int kj = 0; kj < 5; ++kj)
                acc = fmaf(sX[ty + ki][tx + kj], sW[ki * 5 + kj], acc);
        const float base = acc * sSB[0] + sSB[1];

        const int lr = (y >> 1) - cy0, lx = (xg >> 1) - cx0;
        const float L  = sCo[0][lr][lx] + sCo[4][lr][lx];
        const float LH = sCo[1][lr][lx];
        const float HL = sCo[2][lr][lx];
        const float HH = sCo[3][lr][lx];
        const int px = xg & 1, py = y & 1;
        const float v = L + (px ? -LH : LH) + (py ? -HL : HL)
                          + ((px ^ py) ? -HH : HH);
        out[((size_t)(b * C + c) * H + y) * W + xg] = base + 0.5f * v;
    }
}

// ---------------------------------------------------------------------------
// Launch
// ---------------------------------------------------------------------------
extern "C" void kernel_launch(void* const* d_in, const int* in_sizes, int n_in,
                              void* d_out, int out_size, void* d_ws, size_t ws_size,
                              hipStream_t stream) {
    (void)in_sizes; (void)n_in; (void)out_size; (void)ws_size;
    const int B = 8, C = 96;

    const float* x          = (const float*)d_in[0];
    const float* base_w     = (const float*)d_in[1];
    const float* base_b     = (const float*)d_in[2];
    const float* base_scale = (const float*)d_in[3];
    const float* wconv_w    = (const float*)d_in[4];
    const float* wscale     = (const float*)d_in[5];
    float* out = (float*)d_out;
    float* ws  = (float*)d_ws;

    const size_t nBC = (size_t)B * C;
    size_t o = 0;
    float* cur1  = ws + o;  o += nBC * 112 * 112;
    float* cur2  = ws + o;  o += nBC * 56 * 56;
    float* lc0   = ws + o;  o += nBC * 4 * 112 * 112;
    float* lc1   = ws + o;  o += nBC * 4 * 56 * 56;
    float* lc2   = ws + o;  o += nBC * 4 * 28 * 28;
    float* ll56  = ws + o;  o += nBC * 56 * 56;
    float* ll112 = ws + o;  o += nBC * 112 * 112;

    const dim3 blk(256);
    const int W4C25 = 4 * C * 25, W4C = 4 * C;

    { // level 0: 224 -> 112
        const int tiles = 112 / TILE;
        dim3 g(tiles * tiles, C, B);
        k_haar_dwconv<<<g, blk, 0, stream>>>(x, wconv_w + 0 * W4C25, wscale + 0 * W4C,
                                             lc0, cur1, C, 112, 112, tiles);
    }
    { // level 1: 112 -> 56
        const int tiles = 56 / TILE;
        dim3 g(tiles * tiles, C, B);
        k_haar_dwconv<<<g, blk, 0, stream>>>(cur1, wconv_w + 1 * W4C25, wscale + 1 * W4C,
                                             lc1, cur2, C, 56, 56, tiles);
    }
    { // level 2: 56 -> 28
        const int tiles = 28 / TILE;
        dim3 g(tiles * tiles, C, B);
        k_haar_dwconv<<<g, blk, 0, stream>>>(cur2, wconv_w + 2 * W4C25, wscale + 2 * W4C,
                                             lc2, (float*)nullptr, C, 28, 28, tiles);
    }
    { // recon level 2: 28 -> 56
        const int n = (int)(nBC * 28 * 28);
        k_ihaar<<<(n + 255) / 256, blk, 0, stream>>>(lc2, (const float*)nullptr, ll56, n, 28, 28);
    }
    { // recon level 1: 56 -> 112
        const int n = (int)(nBC * 56 * 56);
        k_ihaar<<<(n + 255) / 256, blk, 0, stream>>>(lc1, ll56, ll112, n, 56, 56);
    }
    { // final: base conv + level-0 ihaar + add
        const int tiles = 224 / TILE;
        dim3 g(tiles * tiles, C, B);
        k_final<<<g, blk, 0, stream>>>(x, base_w, base_b, base_scale, lc0, ll112, out, C, tiles);
    }
}